// SpecNorm_29557964931354
// MI455X (gfx1250) — compile-verified
//
#include <hip/hip_runtime.h>
#include <hip/hip_bf16.h>

// SpecNorm EMA-normalization, MI455X (gfx1250) tuned.
//
// Roofline: 123 MB in + 123 MB out => ~10.6 us floor @ 23.3 TB/s. Compute is
// ~0.25 GFLOP total => pure bandwidth problem; matrix ops are irrelevant.
// Strategy: chunked linear scan (EMA is linear => s_end = a^L * s_in + local):
//   pass 1: per (b,f,chunk) chunk-local EMA (also warms the 192 MB L2 with
//           the whole 123 MB input),
//   pass 2: tiny serial combine of chunk carries with alpha^L,
//   pass 3: replay each chunk from its incoming state; input re-read hits L2.
// Pass 3 uses the CDNA5 async DMA path (GLOBAL_LOAD_ASYNC_TO_LDS_B64 +
// ASYNCcnt ring buffer) for an 8-deep, VGPR-free prefetch pipeline over the
// 3848-byte-strided row walk, and NT stores so the output stream does not
// evict the L2-resident input (in + out = 246 MB > 192 MB L2).

typedef __attribute__((ext_vector_type(2))) float f32x2;

#define BB 16
#define TT 2000
#define FF 481
#define CC 40        // chunks per sequence
#define LL 50        // chunk length (CC*LL == TT)
#define BLK 256      // 8 wave32s
#define DPTH 8       // async LDS pipeline depth (power of two)

__host__ __device__ constexpr float alpha_pow(int n) {
    double p = 1.0;
    for (int i = 0; i < n; ++i) p *= 0.9;
    return (float)p;
}

#define ALPHA  0.9f
#define BETA   0.1f            // (1 - ALPHA) in f32, matches jnp f32 math
#define EPS    1e-12f
#define S0_A   0.001f
#define S0_STEP ((0.0001f - 0.001f) / (float)(FF - 1))

// ---------------------------------------------------------------------------
// Pass 1: per (b, f, chunk) chunk-local EMA with s_in = 0.
// Consecutive lanes -> consecutive f -> coalesced 8B/lane (256B/wave) loads.
// Regular-temporal loads intentionally populate L2 (input fits in 192 MB).
// ---------------------------------------------------------------------------
__global__ void specnorm_chunk_carry(const f32x2* __restrict__ x,
                                     float* __restrict__ carry) {
    const int tid = blockIdx.x * blockDim.x + threadIdx.x;
    if (tid >= BB * CC * FF) return;
    const int f  = tid % FF;
    const int bc = tid / FF;
    const int c  = bc % CC;
    const int b  = bc / CC;

    const f32x2* p = x + ((long)(b * TT + c * LL) * FF + f);

    float s = 0.0f;
#pragma unroll 4
    for (int j = 0; j < LL; ++j) {
        // speculative prefetch ~16 rows ahead (global_prefetch_b8; OOB dropped)
        __builtin_prefetch((const void*)(p + (long)(j + 16) * FF), 0, 1);
        f32x2 v = p[(long)j * FF];
        float a = sqrtf(v.x * v.x + v.y * v.y);
        s = ALPHA * s + BETA * a;
    }
    carry[tid] = s;
}

// ---------------------------------------------------------------------------
// Pass 2: per (b, f) serial combine across the CC chunk carries.
// s_in[c+1] = alpha^LL * s_in[c] + carry[c].  7696 threads, 40 steps: ~free.
// ---------------------------------------------------------------------------
__global__ void specnorm_combine(const float* __restrict__ carry,
                                 float* __restrict__ s_in) {
    const int tid = blockIdx.x * blockDim.x + threadIdx.x;
    if (tid >= BB * FF) return;
    const int f = tid % FF;
    const int b = tid / FF;

    const float aL = alpha_pow(LL);
    float s = S0_A + (float)f * S0_STEP;
    for (int c = 0; c < CC; ++c) {
        const int idx = (b * CC + c) * FF + f;
        s_in[idx] = s;
        s = aL * s + carry[idx];
    }
}

// ---------------------------------------------------------------------------
// Pass 3: replay chunk from s_in via an async global->LDS ring buffer.
//
// Ring invariant: prologue issues DPTH async fills; each iteration waits
// ASYNCcnt <= DPTH-1 (async loads complete IN ORDER, so the oldest slot has
// landed), consumes the slot from LDS, then refills that slot for row j+DPTH.
// Tail: refill row index is clamped to LL-1 so exactly one fill is issued per
// iteration and the <=DPTH-1 wait stays sufficient for every iteration; the
// clamped fills land in slots that are never read again (harmless L2 hits).
// ---------------------------------------------------------------------------
__global__ void specnorm_apply(const f32x2* __restrict__ x,
                               const float* __restrict__ s_in,
                               f32x2* __restrict__ out) {
    // slot-major: stage[slot*BLK + lane] => ds_load_b64 is bank-conflict-free
    __shared__ f32x2 stage[DPTH * BLK];   // 16 KB / block (320 KB per WGP)

    const int tid = blockIdx.x * blockDim.x + threadIdx.x;
    if (tid >= BB * CC * FF) return;
    const int f  = tid % FF;
    const int bc = tid / FF;
    const int c  = bc % CC;
    const int b  = bc / CC;

    const long base = (long)(b * TT + c * LL) * FF + f;
    const f32x2* p = x + base;
    f32x2* q = out + base;

    // LDS byte address of this lane's slot 0: low 32 bits of the generic
    // LDS pointer are the LDS byte offset (ISA aperture mapping).
    const unsigned lds0 = (unsigned)(uintptr_t)&stage[threadIdx.x];

    // prologue: fill slots 0..DPTH-1 with rows 0..DPTH-1
#pragma unroll
    for (int k = 0; k < DPTH; ++k) {
        const unsigned laddr = lds0 + (unsigned)(k * BLK * sizeof(f32x2));
        const f32x2* ga = p + (long)k * FF;
        asm volatile("global_load_async_to_lds_b64 %0, %1, off"
                     :: "v"(laddr), "v"(ga) : "memory");
    }

    float s = s_in[tid];
#pragma unroll 4
    for (int j = 0; j < LL; ++j) {
        // oldest outstanding async fill (slot j%DPTH) has completed
        asm volatile("s_wait_asynccnt 0x7" ::: "memory");

        const int slot = j & (DPTH - 1);
        f32x2 v = stage[slot * BLK + threadIdx.x];     // ds_load_b64

        float a = sqrtf(v.x * v.x + v.y * v.y);
        s = ALPHA * s + BETA * a;
        float inv = 1.0f / sqrtf(s + EPS);
        f32x2 o;
        o.x = v.x * inv;
        o.y = v.y * inv;
        __builtin_nontemporal_store(o, q + (long)j * FF);

        // slot consumed (ds_load retired) before we overwrite it
        asm volatile("s_wait_dscnt 0x0" ::: "memory");
        int jn = j + DPTH;
        if (jn > LL - 1) jn = LL - 1;                  // tail clamp (see above)
        const unsigned laddr = lds0 + (unsigned)(slot * BLK * sizeof(f32x2));
        const f32x2* ga = p + (long)jn * FF;
        asm volatile("global_load_async_to_lds_b64 %0, %1, off"
                     :: "v"(laddr), "v"(ga) : "memory");
    }
    // kernel-end implicit S_WAIT_IDLE drains the 8 dangling clamped fills
}

extern "C" void kernel_launch(void* const* d_in, const int* in_sizes, int n_in,
                              void* d_out, int out_size, void* d_ws, size_t ws_size,
                              hipStream_t stream) {
    (void)in_sizes; (void)n_in; (void)out_size; (void)ws_size;

    const f32x2* x = (const f32x2*)d_in[0];
    f32x2* out = (f32x2*)d_out;

    // workspace partition: carry[B*C*F] then s_in[B*C*F]  (2 * 1.23 MB)
    float* carry = (float*)d_ws;
    float* s_in  = carry + (size_t)BB * CC * FF;

    const int nSeqChunks = BB * CC * FF;   // 307,840 threads = 9620 wave32s
    const int nSeqs      = BB * FF;        // 7,696

    specnorm_chunk_carry<<<(nSeqChunks + BLK - 1) / BLK, BLK, 0, stream>>>(x, carry);
    specnorm_combine<<<(nSeqs + BLK - 1) / BLK, BLK, 0, stream>>>(carry, s_in);
    specnorm_apply<<<(nSeqChunks + BLK - 1) / BLK, BLK, 0, stream>>>(x, s_in, out);
}